// GNN4_79783312490855
// MI455X (gfx1250) — compile-verified
//
#include <hip/hip_runtime.h>
#include <hip/hip_bf16.h>

#define N_DRUG 572
#define EMB 256
#define K_NB 128
#define N_EDGE 4576
#define TWO_EMB 512
#define BN_EPS 1e-5f

typedef __attribute__((ext_vector_type(2))) float v2f;
typedef __attribute__((ext_vector_type(8))) float v8f;

__device__ __forceinline__ float wave_sum32(float v) {
#pragma unroll
    for (int off = 16; off > 0; off >>= 1) v += __shfl_xor(v, off, 32);
    return v;
}

// ---------------------------------------------------------------------------
// Kernel 0: zero the atomic-accumulator workspace regions (ws is poisoned).
// ---------------------------------------------------------------------------
__global__ void zero_kernel(float* p, int n) {
    int i = blockIdx.x * blockDim.x + threadIdx.x;
    if (i < n) p[i] = 0.0f;
}

// ---------------------------------------------------------------------------
// Kernel 1: per-drug attention.  One block (256 thr = 8 waves) per drug.
//   scores[k] = <d, rela[adj_rel[n,k]]>  -> softmax -> attended = sum alpha*e
//   X[n] = [attended | d]   (row of the 2E-wide GEMM input)
// Gathers are float4-coalesced: 32 lanes x 8 floats cover one 256-float row.
// ---------------------------------------------------------------------------
__global__ void attn_kernel(const int* __restrict__ drug_name,
                            const int* __restrict__ adj_tail,
                            const int* __restrict__ adj_relation,
                            const float* __restrict__ drug_table,
                            const float* __restrict__ rela_table,
                            const float* __restrict__ ent_table,
                            float* __restrict__ X) {
    const int n    = blockIdx.x;
    const int tid  = threadIdx.x;       // 0..255
    const int lane = tid & 31;
    const int wave = tid >> 5;          // 0..7

    __shared__ __align__(16) float d_sh[EMB];
    __shared__ float sc[K_NB];
    __shared__ int   rel_sh[K_NB];
    __shared__ int   tail_sh[K_NB];

    const int drow = drug_name[n];
    d_sh[tid] = drug_table[(size_t)drow * EMB + tid];
    if (tid < K_NB) {
        rel_sh[tid]  = adj_relation[n * K_NB + tid];
        tail_sh[tid] = adj_tail[n * K_NB + tid];
    }
    __syncthreads();

    // --- scores: each wave owns 16 neighbors ---
    const float4* d4 = (const float4*)d_sh;
    const float4  da = d4[lane * 2 + 0];
    const float4  db = d4[lane * 2 + 1];
#pragma unroll 1
    for (int k = wave; k < K_NB; k += 8) {
        const float4* r4 = (const float4*)(rela_table + (size_t)rel_sh[k] * EMB);
        float4 ra = r4[lane * 2 + 0];
        float4 rb = r4[lane * 2 + 1];
        float p = ra.x*da.x + ra.y*da.y + ra.z*da.z + ra.w*da.w
                + rb.x*db.x + rb.y*db.y + rb.z*db.z + rb.w*db.w;
        p = wave_sum32(p);
        if (lane == 0) sc[k] = p;
    }
    __syncthreads();

    // --- softmax over 128 scores: wave 0, 4 scores per lane ---
    if (wave == 0) {
        float s0 = sc[lane], s1 = sc[lane+32], s2 = sc[lane+64], s3 = sc[lane+96];
        float m = fmaxf(fmaxf(s0, s1), fmaxf(s2, s3));
#pragma unroll
        for (int off = 16; off > 0; off >>= 1) m = fmaxf(m, __shfl_xor(m, off, 32));
        float e0 = expf(s0 - m), e1 = expf(s1 - m), e2 = expf(s2 - m), e3 = expf(s3 - m);
        float inv = 1.0f / wave_sum32(e0 + e1 + e2 + e3);
        sc[lane] = e0*inv; sc[lane+32] = e1*inv; sc[lane+64] = e2*inv; sc[lane+96] = e3*inv;
    }
    __syncthreads();

    // --- attended: thread e owns feature e; ent rows are coalesced.
    //     Prefetch 8 rows ahead (global_prefetch_b8) to hide L2 latency on the
    //     serial FMA-accumulate chain.
    float acc = 0.0f;
#pragma unroll 4
    for (int k = 0; k < K_NB; ++k) {
        if (k + 8 < K_NB)
            __builtin_prefetch(ent_table + (size_t)tail_sh[k + 8] * EMB + tid, 0, 0);
        acc = fmaf(sc[k], ent_table[(size_t)tail_sh[k] * EMB + tid], acc);
    }

    X[n * TWO_EMB + tid]       = acc;       // attended
    X[n * TWO_EMB + EMB + tid] = d_sh[tid]; // d  (concat)
}

// ---------------------------------------------------------------------------
// Kernel 2: H = relu(X @ W + b) via V_WMMA_F32_16X16X4_F32, fused BN stats.
// Each wave computes a 16x64 strip (4 N-tiles) so the A fragment is loaded
// once per k-step and fed to 4 back-to-back WMMAs (A-reuse, XDL-friendly).
// 36 M-tiles x 4 strips = 144 waves = 36 blocks of 4 waves.
// ---------------------------------------------------------------------------
__global__ void gemm_kernel(const float* __restrict__ X,     // [572,512]
                            const float* __restrict__ W,     // [512,256]
                            const float* __restrict__ bias,  // [256]
                            float* __restrict__ H,           // [572,256]
                            float* __restrict__ colsum,      // [256]
                            float* __restrict__ colsumsq) {  // [256]
    const int lane  = threadIdx.x & 31;
    const int wave  = threadIdx.x >> 5;
    const int wtile = blockIdx.x * 4 + wave;  // 0..143
    const int m0 = (wtile >> 2) * 16;         // row tile base (36 tiles)
    const int n0 = (wtile & 3) * 64;          // 64-wide N strip base

    const int half = lane >> 4;               // K-half select (0: K=0..1, 1: K=2..3)
    const int l16  = lane & 15;
    const int arow = m0 + l16;
    const bool arow_ok = arow < N_DRUG;
    const int arow_c = arow_ok ? arow : 0;    // clamped address, value zeroed below

    v8f acc0 = {0.f,0.f,0.f,0.f,0.f,0.f,0.f,0.f};
    v8f acc1 = acc0, acc2 = acc0, acc3 = acc0;

#pragma unroll 4
    for (int k = 0; k < TWO_EMB; k += 4) {
        const int ka = k + half * 2;
        float a0 = X[arow_c * TWO_EMB + ka];
        float a1 = X[arow_c * TWO_EMB + ka + 1];
        a0 = arow_ok ? a0 : 0.0f;             // branchless: EXEC stays all-1s
        a1 = arow_ok ? a1 : 0.0f;
        v2f av = {a0, a1};

        const float* w0 = W + ka * EMB + n0 + l16;        // row ka
        const float* w1 = W + (ka + 1) * EMB + n0 + l16;  // row ka+1
        v2f bv0 = {w0[0],  w1[0]};
        v2f bv1 = {w0[16], w1[16]};
        v2f bv2 = {w0[32], w1[32]};
        v2f bv3 = {w0[48], w1[48]};

        acc0 = __builtin_amdgcn_wmma_f32_16x16x4_f32(false, av, false, bv0, (short)0, acc0, false, false);
        acc1 = __builtin_amdgcn_wmma_f32_16x16x4_f32(false, av, false, bv1, (short)0, acc1, false, false);
        acc2 = __builtin_amdgcn_wmma_f32_16x16x4_f32(false, av, false, bv2, (short)0, acc2, false, false);
        acc3 = __builtin_amdgcn_wmma_f32_16x16x4_f32(false, av, false, bv3, (short)0, acc3, false, false);
    }

    // Epilogue: bias + ReLU + store + per-column BN partial sums.
    v8f* accs[4] = {&acc0, &acc1, &acc2, &acc3};
#pragma unroll
    for (int t = 0; t < 4; ++t) {
        const int col = n0 + t * 16 + l16;
        const float bval = bias[col];
        float s = 0.0f, sq = 0.0f;
        v8f& a = *accs[t];
#pragma unroll
        for (int j = 0; j < 8; ++j) {
            const int m = m0 + j + half * 8;  // C/D layout: M=j (+8 for high lanes)
            float h = fmaxf(a[j] + bval, 0.0f);
            if (m < N_DRUG) {
                H[m * EMB + col] = h;
                s  += h;
                sq += h * h;
            }
        }
        atomicAdd(&colsum[col], s);
        atomicAdd(&colsumsq[col], sq);
    }
}

// ---------------------------------------------------------------------------
// Kernel 3: BatchNorm apply (training-mode batch stats, biased variance).
// ---------------------------------------------------------------------------
__global__ void bn_kernel(const float* __restrict__ H,
                          const float* __restrict__ colsum,
                          const float* __restrict__ colsumsq,
                          const float* __restrict__ gamma,
                          const float* __restrict__ beta,
                          float* __restrict__ out) {
    const int idx = blockIdx.x * blockDim.x + threadIdx.x;
    if (idx >= N_DRUG * EMB) return;
    const int e = idx & (EMB - 1);
    const float invN = 1.0f / (float)N_DRUG;
    const float mean = colsum[e] * invN;
    const float var  = colsumsq[e] * invN - mean * mean;
    out[idx] = (H[idx] - mean) * rsqrtf(var + BN_EPS) * gamma[e] + beta[e];
}

// ---------------------------------------------------------------------------
// Kernel 4: segment-sum scatter over edges (epoch > 1 checked on-device).
// ---------------------------------------------------------------------------
__global__ void scatter_kernel(const int* __restrict__ nb_src,
                               const int* __restrict__ nb_dst,
                               const int* __restrict__ epoch,
                               const float* __restrict__ F,   // = d_out (pre-scatter)
                               float* __restrict__ nb_sum,
                               float* __restrict__ cnt) {
    if (*epoch <= 1) return;
    const int e = blockIdx.x;           // edge id
    const int t = threadIdx.x;          // feature
    const int s = nb_src[e];
    const int d = nb_dst[e];
    atomicAdd(&nb_sum[d * EMB + t], F[s * EMB + t]);
    if (t == 0) atomicAdd(&cnt[d], 1.0f);
}

// ---------------------------------------------------------------------------
// Kernel 5: combine  out = cnt>0 ? 0.5*(nb_sum/max(cnt,1) + out) : out
// ---------------------------------------------------------------------------
__global__ void combine_kernel(const int* __restrict__ epoch,
                               const float* __restrict__ nb_sum,
                               const float* __restrict__ cnt,
                               float* __restrict__ out) {
    if (*epoch <= 1) return;
    const int idx = blockIdx.x * blockDim.x + threadIdx.x;
    if (idx >= N_DRUG * EMB) return;
    const int n = idx >> 8;             // / EMB
    const float c = cnt[n];
    if (c > 0.0f) {
        const float avg = nb_sum[idx] / fmaxf(c, 1.0f);
        out[idx] = (avg + out[idx]) * 0.5f;
    }
}

// ---------------------------------------------------------------------------
extern "C" void kernel_launch(void* const* d_in, const int* in_sizes, int n_in,
                              void* d_out, int out_size, void* d_ws, size_t ws_size,
                              hipStream_t stream) {
    const int*   drug_name    = (const int*)  d_in[0];
    const int*   adj_tail     = (const int*)  d_in[1];
    const int*   adj_relation = (const int*)  d_in[2];
    const int*   nb_src       = (const int*)  d_in[3];
    const int*   nb_dst       = (const int*)  d_in[4];
    const int*   epoch        = (const int*)  d_in[5];
    const float* drug_table   = (const float*)d_in[6];
    const float* rela_table   = (const float*)d_in[7];
    const float* ent_table    = (const float*)d_in[8];
    const float* lin_W        = (const float*)d_in[9];
    const float* lin_b        = (const float*)d_in[10];
    const float* bn_gamma     = (const float*)d_in[11];
    const float* bn_beta      = (const float*)d_in[12];
    float* out = (float*)d_out;

    // Workspace layout (floats)
    float* X        = (float*)d_ws;                 // 572*512
    float* H        = X        + N_DRUG * TWO_EMB;  // 572*256
    float* colsum   = H        + N_DRUG * EMB;      // 256
    float* colsumsq = colsum   + EMB;               // 256
    float* nb_sum   = colsumsq + EMB;               // 572*256
    float* cnt      = nb_sum   + N_DRUG * EMB;      // 572
    const int zero_n = 2 * EMB + N_DRUG * EMB + N_DRUG; // colsum..cnt

    zero_kernel<<<(zero_n + 255) / 256, 256, 0, stream>>>(colsum, zero_n);

    attn_kernel<<<N_DRUG, 256, 0, stream>>>(drug_name, adj_tail, adj_relation,
                                            drug_table, rela_table, ent_table, X);

    // 144 wave-tiles (36 M-tiles x 4 strips of 64 cols), 4 waves/block
    gemm_kernel<<<36, 128, 0, stream>>>(X, lin_W, lin_b, H, colsum, colsumsq);

    const int total = N_DRUG * EMB;
    bn_kernel<<<(total + 255) / 256, 256, 0, stream>>>(H, colsum, colsumsq,
                                                       bn_gamma, bn_beta, out);

    scatter_kernel<<<N_EDGE, EMB, 0, stream>>>(nb_src, nb_dst, epoch, out, nb_sum, cnt);

    combine_kernel<<<(total + 255) / 256, 256, 0, stream>>>(epoch, nb_sum, cnt, out);
}